// GCN_61770219651386
// MI455X (gfx1250) — compile-verified
//
#include <hip/hip_runtime.h>

typedef __attribute__((ext_vector_type(2))) float    v2f;
typedef __attribute__((ext_vector_type(8))) float    v8f;
typedef __attribute__((ext_vector_type(4))) unsigned v4u;
typedef __attribute__((ext_vector_type(8))) int      v8i;
typedef __attribute__((ext_vector_type(4))) int      v4i;

#define NNODES 20000
#define NEDGES 320000
#define NGRAPH 64

__device__ __forceinline__ void atomicAddF(float* p, float v) {
    unsafeAtomicAdd(p, v);   // global_atomic_add_f32
}

// ---------------------------------------------------------------------------
// Tensor Data Mover: 2D tile load Global -> LDS (ISA cdna5 §8 D# layout).
// Tile is tile_w x tile_h elements of 4 bytes, row stride stride0 elements.
// ---------------------------------------------------------------------------
__device__ __forceinline__ void tdm_load_2d_f32(unsigned lds_addr, const void* gptr,
                                                unsigned dim0, unsigned dim1,
                                                unsigned tile_w, unsigned tile_h,
                                                unsigned stride0) {
    unsigned long long ga = (unsigned long long)(uintptr_t)gptr;
    v4u g0;
    g0[0] = 1u;                                          // count=1, no gather
    g0[1] = lds_addr;                                    // LDS byte address
    g0[2] = (unsigned)(ga & 0xFFFFFFFFu);                // global_addr[31:0]
    g0[3] = (unsigned)((ga >> 32) & 0x01FFFFFFu) | (2u << 30);  // addr[56:32] | type=2
    v8i g1;
    g1[0] = (int)(2u << 16);                             // data_size = 4B
    g1[1] = (int)((dim0 & 0xFFFFu) << 16);               // tensor_dim0[15:0]
    g1[2] = (int)(((dim0 >> 16) & 0xFFFFu) | ((dim1 & 0xFFFFu) << 16));
    g1[3] = (int)(((dim1 >> 16) & 0xFFFFu) | ((tile_w & 0xFFFFu) << 16));
    g1[4] = (int)(tile_h & 0xFFFFu);                     // tile_dim1 (tile_dim2=0)
    g1[5] = (int)stride0;                                // tensor_dim0_stride[31:0]
    g1[6] = 0;
    g1[7] = 0;
    v4i gz = {0, 0, 0, 0};
#if __has_include(<hip/amd_detail/amd_gfx1250_TDM.h>)
    v8i gz8 = {};
    __builtin_amdgcn_tensor_load_to_lds(g0, g1, gz, gz, gz8, 0);
#else
    __builtin_amdgcn_tensor_load_to_lds(g0, g1, gz, gz, 0);
#endif
}

// ---------------- degree / dinv ----------------
__global__ void k_deg_init(float* deg, int n) {
    int i = blockIdx.x * blockDim.x + threadIdx.x;
    if (i < n) deg[i] = 1.0f;           // self-loop contributes 1
}

__global__ void k_deg_edges(const int* __restrict__ ei, float* deg, int E) {
    int e = blockIdx.x * blockDim.x + threadIdx.x;
    if (e < E) atomicAddF(&deg[ei[E + e]], 1.0f);   // dst row of edge_index
}

__global__ void k_rsqrt(float* d, int n) {
    int i = blockIdx.x * blockDim.x + threadIdx.x;
    if (i < n) d[i] = rsqrtf(d[i]);
}

// ---------------------------------------------------------------------------
// fp32 WMMA GEMM: C[N,F] = A[N,K] @ W[K,F], K/F compile-time.
// block = 128 threads (4 waves); blockIdx.y = 16-row tile; wave -> 16-col tile.
// A tile (16 x K fp32, contiguous 16*K*4 bytes) DMA'd into LDS by the TDM.
// ---------------------------------------------------------------------------
template <int K, int F>
__global__ __launch_bounds__(128) void gcn_gemm_wmma(const float* __restrict__ A,
                                                     const float* __restrict__ W,
                                                     float* __restrict__ C) {
    __shared__ float ldsA[16 * K];
    const int tid  = threadIdx.x;
    const int lane = tid & 31;
    const int wave = tid >> 5;

    if (wave == 0) {
        // one TDM descriptor per block: linear 16xK fp32 tile
        tdm_load_2d_f32((unsigned)(uintptr_t)(void*)ldsA,
                        A + (size_t)blockIdx.y * 16 * K,
                        /*dim0=*/K, /*dim1=*/16,
                        /*tile_w=*/K, /*tile_h=*/16, /*stride0=*/K);
    }
    __builtin_amdgcn_s_wait_tensorcnt(0);
    __syncthreads();

    // ISA 7.12.2 f32 16x16x4 fragment layouts (wave32):
    //  A: lanes 0-15 -> K=0,1 ; lanes 16-31 -> K=2,3
    //  B: lanes hold N=lane&15; half-waves split K the same way
    const int m    = lane & 15;
    const int koff = (lane >> 4) << 1;
    const int n0   = (blockIdx.x * 4 + wave) * 16;

    const float* ap = ldsA + m * K + koff;
    const float* Wp = W + (size_t)koff * F + n0 + (lane & 15);

    v8f acc = {};
#pragma unroll
    for (int k = 0; k < K; k += 4) {
        v2f a, b;
        a.x = ap[k];
        a.y = ap[k + 1];
        b.x = Wp[(size_t)k * F];
        b.y = Wp[(size_t)(k + 1) * F];
        acc = __builtin_amdgcn_wmma_f32_16x16x4_f32(
            false, a, false, b, (short)0, acc, false, false);
    }

    // C/D layout: VGPR i -> row i (lanes 0-15) / row i+8 (lanes 16-31)
    float* Cp = C + ((size_t)blockIdx.y * 16 + ((lane >> 4) << 3)) * F + n0 + (lane & 15);
#pragma unroll
    for (int i = 0; i < 8; ++i) Cp[(size_t)i * F] = acc[i];
}

// ---------------- out = bias + h * dinv^2 (self loop term) ----------------
__global__ void k_init_out(const float* __restrict__ G, const float* __restrict__ bias,
                           const float* __restrict__ dinv, float* __restrict__ out,
                           int N, int F) {
    int idx = blockIdx.x * blockDim.x + threadIdx.x;    // one float4 each
    if (idx >= N * F / 4) return;
    int base = idx * 4;
    int i = base / F;
    int f = base % F;
    float di = dinv[i];
    float w  = di * di;
    float4 g = *(const float4*)(G + base);
    float4 b = *(const float4*)(bias + f);
    float4 o = { b.x + g.x * w, b.y + g.y * w, b.z + g.z * w, b.w + g.w * w };
    *(float4*)(out + base) = o;
}

// ---------------- edge scatter: out[dst] += h[src] * dinv[s]*dinv[d] -------
__global__ void k_edge_agg(const float* __restrict__ H, const int* __restrict__ ei,
                           const float* __restrict__ dinv, float* __restrict__ out,
                           int E, int F) {
    int e = blockIdx.x * 8 + (threadIdx.x >> 5);
    if (e >= E) return;
    int lane = threadIdx.x & 31;
    int s = ei[e];
    int d = ei[E + e];
    float w = dinv[s] * dinv[d];
    const float* hs = H + (size_t)s * F;
    float* od = out + (size_t)d * F;
    for (int f = lane * 4; f < F; f += 128) {
        float4 v = *(const float4*)(hs + f);
        atomicAddF(od + f + 0, v.x * w);
        atomicAddF(od + f + 1, v.y * w);
        atomicAddF(od + f + 2, v.z * w);
        atomicAddF(od + f + 3, v.w * w);
    }
}

__global__ void k_relu4(float* p, int n4) {
    int idx = blockIdx.x * blockDim.x + threadIdx.x;
    if (idx >= n4) return;
    float4 v = *(float4*)(p + idx * 4);
    v.x = fmaxf(v.x, 0.f); v.y = fmaxf(v.y, 0.f);
    v.z = fmaxf(v.z, 0.f); v.w = fmaxf(v.w, 0.f);
    *(float4*)(p + idx * 4) = v;
}

__global__ void k_zero(float* p, int n) {
    int i = blockIdx.x * blockDim.x + threadIdx.x;
    if (i < n) p[i] = 0.0f;
}

// one wave per node; F == 128
__global__ void k_pool(const float* __restrict__ H, const int* __restrict__ batch,
                       float* __restrict__ psum, float* __restrict__ pcnt, int N) {
    int i = blockIdx.x * 8 + (threadIdx.x >> 5);
    if (i >= N) return;
    int lane = threadIdx.x & 31;
    int b = batch[i];
    float4 v = *(const float4*)(H + (size_t)i * 128 + lane * 4);
    float* ps = psum + (size_t)b * 128 + lane * 4;
    atomicAddF(ps + 0, v.x);
    atomicAddF(ps + 1, v.y);
    atomicAddF(ps + 2, v.z);
    atomicAddF(ps + 3, v.w);
    if (lane == 0) atomicAddF(&pcnt[b], 1.0f);
}

// out[g][j] = mean(psum[g]) @ Wl + bl ; 64 graphs x 2 outputs
__global__ void k_final(const float* __restrict__ psum, const float* __restrict__ pcnt,
                        const float* __restrict__ Wl, const float* __restrict__ bl,
                        float* __restrict__ out) {
    int tid = blockIdx.x * blockDim.x + threadIdx.x;
    if (tid >= NGRAPH * 2) return;
    int g = tid >> 1, j = tid & 1;
    float inv = 1.0f / fmaxf(pcnt[g], 1.0f);
    float acc = bl[j];
    for (int k = 0; k < 128; ++k)
        acc += psum[g * 128 + k] * inv * Wl[k * 2 + j];
    out[g * 2 + j] = acc;
}

extern "C" void kernel_launch(void* const* d_in, const int* in_sizes, int n_in,
                              void* d_out, int out_size, void* d_ws, size_t ws_size,
                              hipStream_t stream) {
    const float* x     = (const float*)d_in[0];
    const int*   ei    = (const int*)  d_in[1];   // [2, E] row-major
    const int*   batch = (const int*)  d_in[2];
    const float* W1    = (const float*)d_in[3];
    const float* b1    = (const float*)d_in[4];
    const float* W2    = (const float*)d_in[5];
    const float* b2    = (const float*)d_in[6];
    const float* W3    = (const float*)d_in[7];
    const float* b3    = (const float*)d_in[8];
    const float* Wl    = (const float*)d_in[9];
    const float* bl    = (const float*)d_in[10];
    float* out = (float*)d_out;

    char* ws = (char*)d_ws;
    size_t off = 0;
    auto carve = [&](size_t bytes) {
        void* p = ws + off;
        off += (bytes + 255) & ~(size_t)255;
        return p;
    };
    float* dinv = (float*)carve((size_t)NNODES * 4);
    float* bufA = (float*)carve((size_t)NNODES * 512 * 4);   // gemm outputs
    float* bufB = (float*)carve((size_t)NNODES * 512 * 4);   // aggregated outputs
    float* psum = (float*)carve((size_t)(NGRAPH * 128 + NGRAPH) * 4);
    float* pcnt = psum + NGRAPH * 128;

    const int N = NNODES, E = NEDGES;

    k_deg_init<<<(N + 255) / 256, 256, 0, stream>>>(dinv, N);
    k_deg_edges<<<(E + 255) / 256, 256, 0, stream>>>(ei, dinv, E);
    k_rsqrt<<<(N + 255) / 256, 256, 0, stream>>>(dinv, N);

    auto post = [&](int F, const float* b, bool relu) {
        k_init_out<<<(N * F / 4 + 255) / 256, 256, 0, stream>>>(bufA, b, dinv, bufB, N, F);
        k_edge_agg<<<(E + 7) / 8, 256, 0, stream>>>(bufA, ei, dinv, bufB, E, F);
        if (relu)
            k_relu4<<<(N * F / 4 + 255) / 256, 256, 0, stream>>>(bufB, N * F / 4);
    };

    // layer 1: [N,128] @ [128,512]
    gcn_gemm_wmma<128, 512><<<dim3(512 / 64, N / 16), 128, 0, stream>>>(x, W1, bufA);
    post(512, b1, true);
    // layer 2: [N,512] @ [512,256]
    gcn_gemm_wmma<512, 256><<<dim3(256 / 64, N / 16), 128, 0, stream>>>(bufB, W2, bufA);
    post(256, b2, true);
    // layer 3: [N,256] @ [256,128]
    gcn_gemm_wmma<256, 128><<<dim3(128 / 64, N / 16), 128, 0, stream>>>(bufB, W3, bufA);
    post(128, b3, false);

    k_zero<<<(NGRAPH * 128 + NGRAPH + 255) / 256, 256, 0, stream>>>(psum, NGRAPH * 128 + NGRAPH);
    k_pool<<<(N + 7) / 8, 256, 0, stream>>>(bufB, batch, psum, pcnt, N);
    k_final<<<1, 128, 0, stream>>>(psum, pcnt, Wl, bl, out);
}